// LRUCell_63221918597687
// MI455X (gfx1250) — compile-verified
//
#include <hip/hip_runtime.h>
#include <math.h>

// LRU cell: GEMM(WMMA f32) -> chunked complex scan -> GEMM(WMMA f32)+ReLU
// Shapes: B=8, L=4096, D=512 (complex channels), 2D=1024.

typedef __attribute__((ext_vector_type(2))) float v2f;
typedef __attribute__((ext_vector_type(8))) float v8f;

#define B_      8
#define L_      4096
#define D_      512
#define TWO_D   1024
#define NCHUNK  16
#define CHUNK   (L_ / NCHUNK)   // 256
#define MTOT    (B_ * L_)       // 32768

// ---------------------------------------------------------------------------
// GEMM: C[M x N] = A[M x K] * W[N x K]^T + bias, optional ReLU.
// Workgroup tile 128(M) x 64(N), 4 wave32 waves (each 32x64), K staged 32 at
// a time through double-buffered LDS so global loads overlap the WMMA stream.
// Uses V_WMMA_F32_16X16X4_F32 (full fp32 precision, matches fp32 reference).
// ---------------------------------------------------------------------------
template <int K, bool RELU>
__global__ __launch_bounds__(128) void gemm_wmma_f32(
    const float* __restrict__ A, const float* __restrict__ W,
    const float* __restrict__ bias, float* __restrict__ C, int Ntotal)
{
  static_assert(K % 32 == 0, "K must be a multiple of 32");
  constexpr int NC = K / 32;

  // pitch 36 floats: rows stay 16B-aligned for b128 LDS ops, conflict-free reads
  __shared__ float lA[2][128][36];   // 36 KB
  __shared__ float lW[2][64][36];    // 18 KB

  const int tid  = threadIdx.x;
  const int lane = tid & 31;
  const int wv   = tid >> 5;          // wave id 0..3  -> 32-row M sub-tile
  const int n0   = blockIdx.x * 64;
  const int m0   = blockIdx.y * 128;

  const int lrow  = lane & 15;        // row within 16-lane half
  const int khalf = (lane >> 4) * 2;  // lanes 16..31 hold K pair {2,3}

  v8f acc[2][4];
#pragma unroll
  for (int ms = 0; ms < 2; ++ms)
#pragma unroll
    for (int t = 0; t < 4; ++t) acc[ms][t] = (v8f){};

  // cooperative stage of A(128x32) and W(64x32) tiles into LDS buffer `buf`
  auto stage = [&](int c, int buf) {
    const int k0 = c * 32;
#pragma unroll
    for (int i = tid; i < 1024; i += 128) {        // A: 1024 float4
      const int row = i >> 3;                      // 0..127
      const int col = (i & 7) << 2;                // 0,4,...,28
      *(float4*)&lA[buf][row][col] =
          *(const float4*)(A + (size_t)(m0 + row) * K + k0 + col);
    }
#pragma unroll
    for (int i = tid; i < 512; i += 128) {         // W: 512 float4
      const int row = i >> 3;                      // 0..63
      const int col = (i & 7) << 2;
      *(float4*)&lW[buf][row][col] =
          *(const float4*)(W + (size_t)(n0 + row) * K + k0 + col);
    }
  };

  stage(0, 0);
  __syncthreads();

  for (int c = 0; c < NC; ++c) {
    const int buf = c & 1;
    if (c + 1 < NC) stage(c + 1, buf ^ 1);         // overlap next tile's loads
    if (c + 2 < NC) {                              // L2 prefetch 2 tiles ahead
      __builtin_prefetch(A + (size_t)(m0 + (tid & 127)) * K + (c + 2) * 32, 0, 3);
      __builtin_prefetch(W + (size_t)(n0 + (tid & 63)) * K + (c + 2) * 32, 0, 3);
    }

    // ---- 8 K-steps of 4: 6 LDS fragments feed 8 WMMAs per step ----
#pragma unroll
    for (int kk = 0; kk < 32; kk += 4) {
      v2f a0, a1;  // A fragments: 16x4, lane layout per ISA 7.12.2
      a0.x = lA[buf][32 * wv + lrow][kk + khalf];
      a0.y = lA[buf][32 * wv + lrow][kk + khalf + 1];
      a1.x = lA[buf][32 * wv + 16 + lrow][kk + khalf];
      a1.y = lA[buf][32 * wv + 16 + lrow][kk + khalf + 1];
#pragma unroll
      for (int t = 0; t < 4; ++t) {
        v2f bb;  // B fragment: 4x16, B[k][n] = W[n][k]
        bb.x = lW[buf][16 * t + lrow][kk + khalf];
        bb.y = lW[buf][16 * t + lrow][kk + khalf + 1];
        acc[0][t] = __builtin_amdgcn_wmma_f32_16x16x4_f32(
            false, a0, false, bb, (short)0, acc[0][t], false, false);
        acc[1][t] = __builtin_amdgcn_wmma_f32_16x16x4_f32(
            false, a1, false, bb, (short)0, acc[1][t], false, false);
      }
    }
    __syncthreads();
  }

  // ---- store: C/D layout = VGPR r, lanes 0..15 -> M=r, lanes 16..31 -> M=r+8
  const int lhi = lane >> 4;
#pragma unroll
  for (int t = 0; t < 4; ++t) {
    const int ncol = n0 + 16 * t + lrow;
    const float bv = bias[ncol];
#pragma unroll
    for (int ms = 0; ms < 2; ++ms) {
#pragma unroll
      for (int r = 0; r < 8; ++r) {
        const int mrow = m0 + 32 * wv + 16 * ms + r + 8 * lhi;
        float val = acc[ms][t][r] + bv;
        if (RELU) val = fmaxf(val, 0.0f);
        C[(size_t)mrow * Ntotal + ncol] = val;
      }
    }
  }
}

// ---------------------------------------------------------------------------
// Wo column repack: Wo2[o][2d] = Wo[o][d], Wo2[o][2d+1] = Wo[o][512+d]
// lets GEMM2 consume the interleaved (re,im) state layout directly.
// ---------------------------------------------------------------------------
__global__ void repack_wo(const float* __restrict__ Wo, float* __restrict__ Wo2)
{
  const int i = blockIdx.x * blockDim.x + threadIdx.x;
  if (i >= D_ * TWO_D) return;
  const int o    = i >> 10;     // output row
  const int c    = i & 1023;    // packed column
  const int d    = c >> 1;
  const int part = c & 1;       // 0 = re, 1 = im
  Wo2[i] = Wo[(size_t)o * TWO_D + part * D_ + d];
}

// ---------------------------------------------------------------------------
// Scan phase A: per-chunk local scans h_t = lamb*h_{t-1} + x_t (in place).
// One thread per complex channel; grid = B * NCHUNK blocks of D threads.
// ---------------------------------------------------------------------------
__global__ __launch_bounds__(D_) void scan_local(
    float* __restrict__ u, const float* __restrict__ plog,
    float2* __restrict__ E)
{
  const int d     = threadIdx.x;
  const int chunk = blockIdx.x % NCHUNK;
  const int b     = blockIdx.x / NCHUNK;

  const float v   = expf(plog[d]);
  const float th  = expf(plog[D_ + d]);
  const float mag = expf(-v);
  const float lre = mag * cosf(th);
  const float lim = mag * sinf(th);

  float2* base = (float2*)u + ((size_t)b * L_ + (size_t)chunk * CHUNK) * D_ + d;

  float hr = 0.0f, hi = 0.0f;
  for (int t = 0; t < CHUNK; ++t) {
    const float2 x = base[(size_t)t * D_];
    const float nr = fmaf(lre, hr, fmaf(-lim, hi, x.x));
    const float ni = fmaf(lre, hi, fmaf( lim, hr, x.y));
    hr = nr; hi = ni;
    base[(size_t)t * D_] = make_float2(hr, hi);
  }
  E[((size_t)b * NCHUNK + chunk) * D_ + d] = make_float2(hr, hi);
}

// ---------------------------------------------------------------------------
// Scan phase B: 16-step cross-chunk carry scan with lamb^CHUNK (double-prec
// power to keep reference-level accuracy). P[c] = full state at end of chunk
// c-1 (P[0] = 0).
// ---------------------------------------------------------------------------
__global__ __launch_bounds__(D_) void scan_carry(
    const float2* __restrict__ E, float2* __restrict__ P,
    const float* __restrict__ plog)
{
  const int d = threadIdx.x;
  const int b = blockIdx.x;

  const float v  = expf(plog[d]);
  const float th = expf(plog[D_ + d]);
  const double magC = exp(-(double)CHUNK * (double)v);
  const double angC = (double)CHUNK * (double)th;
  const float lcr = (float)(magC * cos(angC));
  const float lci = (float)(magC * sin(angC));

  float hr = 0.0f, hi = 0.0f;
  for (int c = 0; c < NCHUNK; ++c) {
    P[((size_t)b * NCHUNK + c) * D_ + d] = make_float2(hr, hi);
    const float2 e = E[((size_t)b * NCHUNK + c) * D_ + d];
    const float nr = e.x + lcr * hr - lci * hi;
    const float ni = e.y + lcr * hi + lci * hr;
    hr = nr; hi = ni;
  }
}

// ---------------------------------------------------------------------------
// Scan phase C: h_t += lamb^{t+1} * P[chunk], then scale by gamma (in place).
// ---------------------------------------------------------------------------
__global__ __launch_bounds__(D_) void scan_fix(
    float* __restrict__ u, const float* __restrict__ plog,
    const float2* __restrict__ P)
{
  const int d     = threadIdx.x;
  const int chunk = blockIdx.x % NCHUNK;
  const int b     = blockIdx.x / NCHUNK;

  const float v     = expf(plog[d]);
  const float th    = expf(plog[D_ + d]);
  const float gamma = expf(plog[2 * D_ + d]);
  const float mag   = expf(-v);
  const float lre   = mag * cosf(th);
  const float lim   = mag * sinf(th);

  const float2 p = P[((size_t)b * NCHUNK + chunk) * D_ + d];
  float wr = lre * p.x - lim * p.y;   // w = lamb^1 * P
  float wi = lre * p.y + lim * p.x;

  float2* base = (float2*)u + ((size_t)b * L_ + (size_t)chunk * CHUNK) * D_ + d;
  for (int t = 0; t < CHUNK; ++t) {
    const float2 x = base[(size_t)t * D_];
    base[(size_t)t * D_] = make_float2((x.x + wr) * gamma, (x.y + wi) * gamma);
    const float nwr = lre * wr - lim * wi;
    const float nwi = lre * wi + lim * wr;
    wr = nwr; wi = nwi;
  }
}

// ---------------------------------------------------------------------------
extern "C" void kernel_launch(void* const* d_in, const int* in_sizes, int n_in,
                              void* d_out, int out_size, void* d_ws, size_t ws_size,
                              hipStream_t stream)
{
  const float* inputs = (const float*)d_in[0];   // (8,4096,512)
  const float* Wi     = (const float*)d_in[1];   // (1024,512)
  const float* bi     = (const float*)d_in[2];   // (1024,)
  const float* Wo     = (const float*)d_in[3];   // (512,1024)
  const float* bo     = (const float*)d_in[4];   // (512,)
  const float* plog   = (const float*)d_in[5];   // (3,512)
  float* out          = (float*)d_out;           // (8,4096,512)

  // workspace layout (needs ~131 MB; u fits entirely in MI455X's 192 MB L2)
  char* ws = (char*)d_ws;
  float*  u   = (float*)ws;                                   // 32768*1024 f32
  float*  Wo2 = (float*)(ws + (size_t)MTOT * TWO_D * 4);      // 512*1024 f32
  float2* E   = (float2*)((char*)Wo2 + (size_t)D_ * TWO_D * 4);
  float2* P   = (float2*)((char*)E + (size_t)B_ * NCHUNK * D_ * sizeof(float2));

  // 1) u = inputs @ Wi^T + bi           (WMMA f32, M=32768 N=1024 K=512)
  dim3 g1(TWO_D / 64, MTOT / 128);
  gemm_wmma_f32<D_, false><<<g1, 128, 0, stream>>>(inputs, Wi, bi, u, TWO_D);

  // 2) repack Wo for the interleaved (re,im) state layout
  repack_wo<<<(D_ * TWO_D) / 256, 256, 0, stream>>>(Wo, Wo2);

  // 3) chunked complex scan (A: local, B: carries, C: fix-up + gamma)
  scan_local<<<B_ * NCHUNK, D_, 0, stream>>>(u, plog, E);
  scan_carry<<<B_, D_, 0, stream>>>(E, P, plog);
  scan_fix<<<B_ * NCHUNK, D_, 0, stream>>>(u, plog, P);

  // 4) out = relu(x @ Wo2^T + bo)       (WMMA f32, M=32768 N=512 K=1024)
  dim3 g2(D_ / 64, MTOT / 128);
  gemm_wmma_f32<TWO_D, true><<<g2, 128, 0, stream>>>(u, Wo2, bo, out, D_);
}